// SelfAttention_12824772345986
// MI455X (gfx1250) — compile-verified
//
#include <hip/hip_runtime.h>
#include <math.h>

typedef __attribute__((ext_vector_type(2))) float v2f;
typedef __attribute__((ext_vector_type(4))) float v4f;
typedef __attribute__((ext_vector_type(8))) float v8f;

#define TT 2048
#define DD 1024
#define HH 16
#define DHH 64

__device__ __forceinline__ v8f wmma_f32(v2f a, v2f b, v8f c) {
  // D = A(16x4) * B(4x16) + C, fp32 throughout
  return __builtin_amdgcn_wmma_f32_16x16x4_f32(false, a, false, b, (short)0, c, false, false);
}

__device__ __forceinline__ v8f zero8() {
  v8f z;
#pragma unroll
  for (int i = 0; i < 8; i++) z[i] = 0.0f;
  return z;
}

// ---------------------------------------------------------------------------
// Kernel 1: absolute position bias AP[t,s]:
//   i = s/2 ; angle = t * 10000^(-2i/D) ; even s -> sin, odd s -> cos
// ---------------------------------------------------------------------------
__global__ void ap_kernel(float* __restrict__ AP) {
  int idx = blockIdx.x * 256 + threadIdx.x;       // covers T*T
  int t = idx >> 11;
  int s = idx & (TT - 1);
  int i = s >> 1;
  // 10000^(-2i/1024) = exp2(-log2(10000) * 2i / 1024)
  float angle = (float)t * exp2f(-13.287712379549449f * (float)(2 * i) * (1.0f / 1024.0f));
  AP[idx] = (s & 1) ? cosf(angle) : sinf(angle);
}

// ---------------------------------------------------------------------------
// Kernel 2/5: tiled fp32 WMMA GEMM, BM=128 BN=64 BK=16, 256 threads (8 waves).
// BMODE: 1 -> B element (k,n) at B[n*ldb + k]        (transposed B, for Wout)
//        2 -> B element (k,n) at B[(n>>6)*Kd*64 + k*64 + (n&63)]  (headed W[H][D][dh])
// CMODE: 0 -> C[m*ldc + n] ; 1 -> C[(n>>6)*M*64 + m*64 + (n&63)]  (headed [H][T][dh])
// ---------------------------------------------------------------------------
template <int BMODE, int CMODE>
__global__ __launch_bounds__(256) void gemm128x64(
    const float* __restrict__ A, const float* __restrict__ B, float* __restrict__ C,
    int M, int N, int Kd, int lda, int ldb, int ldc) {
  __shared__ float As[128][20];   // [m][k], padded: float4 stores & float2 reads aligned
  __shared__ float Bs[64][20];    // [n][k]

  const int tid = threadIdx.x;
  const int lane = tid & 31;
  const int wid = tid >> 5;
  const int m0 = blockIdx.y * 128;
  const int n0 = blockIdx.x * 64;
  const int wm = (wid >> 1) * 32;       // wave tile origin in M (4 rows of waves)
  const int wn = (wid & 1) * 32;        // wave tile origin in N (2 cols of waves)
  const int lm = lane & 15;
  const int kb = (lane >> 4) << 1;      // K sub-offset per lane group

  v8f acc[2][2];
#pragma unroll
  for (int i = 0; i < 2; i++)
#pragma unroll
    for (int j = 0; j < 2; j++) acc[i][j] = zero8();

  for (int k0 = 0; k0 < Kd; k0 += 16) {
    // --- stage A tile: 128x16 floats = 512 float4, 2 per thread ---
#pragma unroll
    for (int i = 0; i < 2; i++) {
      int idx = tid + i * 256;
      int r = idx >> 2;
      int c4 = (idx & 3) << 2;
      v4f av = *(const v4f*)&A[(size_t)(m0 + r) * lda + k0 + c4];
      *(v4f*)&As[r][c4] = av;
    }
    // --- stage B tile: 16x64 floats = 256 float4, 1 per thread ---
    if (BMODE == 1) {
      int r = tid >> 2;                 // n within tile
      int c4 = (tid & 3) << 2;          // k within tile
      v4f bv = *(const v4f*)&B[(size_t)(n0 + r) * ldb + k0 + c4];
      *(v4f*)&Bs[r][c4] = bv;
    } else {
      int r = tid >> 4;                 // k within tile
      int c4 = (tid & 15) << 2;         // n within tile
      const float* bp = &B[(size_t)(n0 >> 6) * Kd * 64 + (size_t)(k0 + r) * 64 + c4];
      v4f bv = *(const v4f*)bp;
#pragma unroll
      for (int j = 0; j < 4; j++) Bs[c4 + j][r] = bv[j];
    }
    __syncthreads();

#pragma unroll
    for (int kk = 0; kk < 16; kk += 4) {
      v2f a0 = *(const v2f*)&As[wm + lm][kk + kb];
      v2f a1 = *(const v2f*)&As[wm + 16 + lm][kk + kb];
      v2f b0 = *(const v2f*)&Bs[wn + lm][kk + kb];
      v2f b1 = *(const v2f*)&Bs[wn + 16 + lm][kk + kb];
      acc[0][0] = wmma_f32(a0, b0, acc[0][0]);
      acc[0][1] = wmma_f32(a0, b1, acc[0][1]);
      acc[1][0] = wmma_f32(a1, b0, acc[1][0]);
      acc[1][1] = wmma_f32(a1, b1, acc[1][1]);
    }
    __syncthreads();
  }

  const int g = lane >> 4;
#pragma unroll
  for (int i = 0; i < 2; i++)
#pragma unroll
    for (int j = 0; j < 2; j++) {
      int n = n0 + wn + j * 16 + lm;
#pragma unroll
      for (int r = 0; r < 8; r++) {
        int m = m0 + wm + i * 16 + r + 8 * g;
        float v = acc[i][j][r];
        if (CMODE == 1)
          C[(size_t)(n >> 6) * M * 64 + (size_t)m * 64 + (n & 63)] = v;
        else
          C[(size_t)m * ldc + n] = v;
      }
    }
}

// ---------------------------------------------------------------------------
// Kernel 3: flash attention, one (head, 64-query-row block) per workgroup.
// 256 threads = 8 waves; wave w owns S/O tile rows ti=w>>1, cols (w&1)*2+{0,1}.
// ---------------------------------------------------------------------------
__global__ __launch_bounds__(256) void attn_flash(
    const float* __restrict__ Q, const float* __restrict__ K, const float* __restrict__ V,
    const float* __restrict__ AP, float* __restrict__ Y,
    float* __restrict__ m15, float* __restrict__ l15) {
  __shared__ float Ss[64][66];          // raw scores, then probabilities P
  __shared__ float mrow[64], lrow[64], fac[64];
  __shared__ float red[64][4], sred[64][4];

  const int tid = threadIdx.x;
  const int lane = tid & 31;
  const int wid = tid >> 5;
  const int h = blockIdx.y;
  const int q0 = blockIdx.x * 64;
  const int ti = wid >> 1;
  const int tj2 = (wid & 1) * 2;
  const int lm = lane & 15;
  const int g = lane >> 4;
  const int kb = g * 2;

  const float* Qh = Q + (size_t)h * TT * DHH;
  const float* Kh = K + (size_t)h * TT * DHH;
  const float* Vh = V + (size_t)h * TT * DHH;

  // Q fragments for this wave's 16 query rows: kept in registers all key blocks.
  v2f aq[16];
#pragma unroll
  for (int kk = 0; kk < 16; kk++)
    aq[kk] = *(const v2f*)&Qh[(size_t)(q0 + ti * 16 + lm) * DHH + kk * 4 + kb];

  v8f oacc[2];
  oacc[0] = zero8();
  oacc[1] = zero8();

  if (tid < 64) { mrow[tid] = -INFINITY; lrow[tid] = 0.0f; }
  __syncthreads();

  const int row = tid >> 2;   // softmax: 4 threads per score row
  const int part = tid & 3;

  for (int s0 = 0; s0 < TT; s0 += 64) {
    // ---- S = Q Kt (two 16x16 tiles per wave, K loop = 64) ----
#pragma unroll
    for (int jj = 0; jj < 2; jj++) {
      int tj = tj2 + jj;
      v8f sacc = zero8();
#pragma unroll
      for (int kk = 0; kk < 16; kk++) {
        v2f b = *(const v2f*)&Kh[(size_t)(s0 + tj * 16 + lm) * DHH + kk * 4 + kb];
        sacc = wmma_f32(aq[kk], b, sacc);
      }
#pragma unroll
      for (int r = 0; r < 8; r++)
        Ss[ti * 16 + r + 8 * g][tj * 16 + lm] = sacc[r];
    }
    __syncthreads();

    // ---- online softmax update (add AP bias here, coalesced) ----
    float sv[16];
    float pm = -INFINITY;
    const float* aprow = &AP[(size_t)(q0 + row) * TT + s0 + part * 16];
#pragma unroll
    for (int c = 0; c < 16; c++) {
      float v = Ss[row][part * 16 + c] + aprow[c];
      sv[c] = v;
      pm = fmaxf(pm, v);
    }
    red[row][part] = pm;
    __syncthreads();
    float mnew = fmaxf(fmaxf(red[row][0], red[row][1]), fmaxf(red[row][2], red[row][3]));
    mnew = fmaxf(mnew, mrow[row]);
    float ps = 0.0f;
#pragma unroll
    for (int c = 0; c < 16; c++) {
      float e = __expf(sv[c] - mnew);
      Ss[row][part * 16 + c] = e;
      ps += e;
    }
    sred[row][part] = ps;
    __syncthreads();
    if (part == 0) {
      float f = __expf(mrow[row] - mnew);
      lrow[row] = lrow[row] * f + sred[row][0] + sred[row][1] + sred[row][2] + sred[row][3];
      mrow[row] = mnew;
      fac[row] = f;
    }
    __syncthreads();

    // ---- rescale O, then O += P @ V ----
#pragma unroll
    for (int jj = 0; jj < 2; jj++) {
#pragma unroll
      for (int r = 0; r < 8; r++) oacc[jj][r] *= fac[ti * 16 + r + 8 * g];
#pragma unroll
      for (int kk = 0; kk < 16; kk++) {
        v2f a = *(const v2f*)&Ss[ti * 16 + lm][kk * 4 + kb];
        v2f b;
        b[0] = Vh[(size_t)(s0 + kk * 4 + kb) * DHH + (tj2 + jj) * 16 + lm];
        b[1] = Vh[(size_t)(s0 + kk * 4 + kb + 1) * DHH + (tj2 + jj) * 16 + lm];
        oacc[jj] = wmma_f32(a, b, oacc[jj]);
      }
    }
    __syncthreads();
  }

  // ---- normalize by l and write Y in concat-head layout [T][H*dh] ----
  if (tid < 64) fac[tid] = 1.0f / lrow[tid];
  __syncthreads();
#pragma unroll
  for (int jj = 0; jj < 2; jj++) {
    int n = h * DHH + (tj2 + jj) * 16 + lm;
#pragma unroll
    for (int r = 0; r < 8; r++) {
      int m = q0 + ti * 16 + r + 8 * g;
      Y[(size_t)m * DD + n] = oacc[jj][r] * fac[ti * 16 + r + 8 * g];
    }
  }
  if (h == HH - 1 && tid < 64) {
    m15[q0 + tid] = mrow[tid];
    l15[q0 + tid] = lrow[tid];
  }
}

// ---------------------------------------------------------------------------
// Kernel 4: recompute head-15 scores and emit normalized attention probs.
// ---------------------------------------------------------------------------
__global__ __launch_bounds__(256) void attn_probs(
    const float* __restrict__ Q15, const float* __restrict__ K15,
    const float* __restrict__ AP, const float* __restrict__ m15,
    const float* __restrict__ l15, float* __restrict__ out) {
  const int tid = threadIdx.x;
  const int lane = tid & 31;
  const int wid = tid >> 5;
  const int ti = wid >> 1;
  const int tj2 = (wid & 1) * 2;
  const int lm = lane & 15;
  const int g = lane >> 4;
  const int kb = g * 2;
  const int q0 = blockIdx.y * 64;
  const int s0 = blockIdx.x * 64;

  v2f aq[16];
#pragma unroll
  for (int kk = 0; kk < 16; kk++)
    aq[kk] = *(const v2f*)&Q15[(size_t)(q0 + ti * 16 + lm) * DHH + kk * 4 + kb];

#pragma unroll
  for (int jj = 0; jj < 2; jj++) {
    int tj = tj2 + jj;
    v8f sacc = zero8();
#pragma unroll
    for (int kk = 0; kk < 16; kk++) {
      v2f b = *(const v2f*)&K15[(size_t)(s0 + tj * 16 + lm) * DHH + kk * 4 + kb];
      sacc = wmma_f32(aq[kk], b, sacc);
    }
#pragma unroll
    for (int r = 0; r < 8; r++) {
      int m = q0 + ti * 16 + r + 8 * g;
      int n = s0 + tj * 16 + lm;
      float sval = sacc[r] + AP[(size_t)m * TT + n];
      out[(size_t)m * TT + n] = __expf(sval - m15[m]) / l15[m];
    }
  }
}

// ---------------------------------------------------------------------------
// Host-side launch
// ---------------------------------------------------------------------------
extern "C" void kernel_launch(void* const* d_in, const int* in_sizes, int n_in,
                              void* d_out, int out_size, void* d_ws, size_t ws_size,
                              hipStream_t stream) {
  (void)in_sizes; (void)n_in; (void)out_size; (void)ws_size;
  const float* x    = (const float*)d_in[0];   // [T,D]
  const float* Wq   = (const float*)d_in[1];   // [H,D,dh]
  const float* Wk   = (const float*)d_in[2];
  const float* Wv   = (const float*)d_in[3];
  const float* Wout = (const float*)d_in[4];   // [D,D]

  float* y        = (float*)d_out;                 // [T,D]
  float* attn_out = y + (size_t)TT * DD;           // [T,T]

  float* ws = (float*)d_ws;
  float* Q   = ws;                                 // [H,T,dh]
  float* K   = Q + (size_t)HH * TT * DHH;
  float* V   = K + (size_t)HH * TT * DHH;
  float* Y   = V + (size_t)HH * TT * DHH;          // [T, H*dh]
  float* AP  = Y + (size_t)TT * DD;                // [T,T]
  float* m15 = AP + (size_t)TT * TT;
  float* l15 = m15 + TT;

  // 1) position bias
  ap_kernel<<<(TT * TT) / 256, 256, 0, stream>>>(AP);

  // 2) Q/K/V projections: x[T,D] @ W[H,D,dh] -> [H,T,dh]
  dim3 gproj(DD / 64, TT / 128);
  gemm128x64<2, 1><<<gproj, 256, 0, stream>>>(x, Wq, Q, TT, DD, DD, DD, 0, 0);
  gemm128x64<2, 1><<<gproj, 256, 0, stream>>>(x, Wk, K, TT, DD, DD, DD, 0, 0);
  gemm128x64<2, 1><<<gproj, 256, 0, stream>>>(x, Wv, V, TT, DD, DD, DD, 0, 0);

  // 3) flash attention per head, writes concat-head Y and head-15 m/l
  attn_flash<<<dim3(TT / 64, HH), 256, 0, stream>>>(Q, K, V, AP, Y, m15, l15);

  // 4) head-15 attention probabilities
  attn_probs<<<dim3(TT / 64, TT / 64), 256, 0, stream>>>(
      Q + (size_t)(HH - 1) * TT * DHH, K + (size_t)(HH - 1) * TT * DHH,
      AP, m15, l15, attn_out);

  // 5) out projection: y = Y @ Wout^T  (B accessed as Wout[n*D + k])
  gemm128x64<1, 0><<<dim3(DD / 64, TT / 128), 256, 0, stream>>>(
      Y, Wout, y, TT, DD, DD, DD, DD, DD);
}